// ALIFSpikes_47579647705427
// MI455X (gfx1250) — compile-verified
//
#include <hip/hip_runtime.h>

// ALIF spikes + delay-buffer gather, fused single pass.
// Output layout (1+ND+NDX+1, B, N): [X | Xd(ND) | Xd_xarea(NDX) | new_threshold]
//   slice 0        : X = (V - (thr+1) >= 0) ? 1 : 0
//   slice 1+k      : delays[k]==0 ? X : buffer[delays[k]-1]   (k < ND)
//   slice 1+ND+k   : delays_xarea[k]==0 ? X : buffer[...]     (k < NDX)
//   slice 1+ND+NDX : thr*alpha[n] + X*amplitude[n]
//
// Memory-bound streaming kernel (~480 MB per launch -> ~20 us floor @ 23.3 TB/s).
// No matrix math -> no WMMA; the win is in the data-movement path:
//   * 2D grid (b, n4): n-index comes straight from blockIdx.x, so there is NO
//     integer div/mod on the hot path (previous version burned ~15 SALU/VALU
//     instructions per thread on a magic-number division).
//   * B128 (float4) fully-coalesced loads/stores: 512 B per wave32 access.
//   * Non-temporal stores for all output slices (written once, never re-read).
//   * Delay values pulled through v_readfirstlane -> scalar branch; the
//     buffer load is skipped wave-wide when the slice is just X.
//   * global_prefetch_b8 stream-ahead on the two hot input streams.

typedef float f32x4 __attribute__((ext_vector_type(4)));

#define BLOCK 256

template <int ND_C, int NDX_C>
__global__ __launch_bounds__(BLOCK)
void alif_fused_kernel(const f32x4* __restrict__ V4,
                       const f32x4* __restrict__ T4,
                       const f32x4* __restrict__ A4,   // alpha,  (N/4) vec4
                       const f32x4* __restrict__ AM4,  // amplitude
                       const f32x4* __restrict__ BUF4, // (DMAX, B, N)
                       const int*   __restrict__ delays,
                       const int*   __restrict__ delays_x,
                       f32x4*       __restrict__ OUT4, // (1+ND+NDX+1, B, N)
                       int N4, int BN4, int nd_rt, int ndx_rt)
{
    const int nd  = (ND_C  >= 0) ? ND_C  : nd_rt;
    const int ndx = (NDX_C >= 0) ? NDX_C : ndx_rt;

    const int n4 = blockIdx.x * BLOCK + threadIdx.x;   // vec4 column in a row
    if (n4 >= N4) return;
    const int i  = blockIdx.y * N4 + n4;               // flat vec4 index in (B,N)

    // Stream-ahead prefetch of the two hot input streams (global_prefetch_b8).
    const int PFD = 4096;                               // 64 KB ahead (vec4 units)
    if (i + PFD < BN4) {
        __builtin_prefetch((const void*)(V4 + i + PFD), 0, 0);
        __builtin_prefetch((const void*)(T4 + i + PFD), 0, 0);
    }

    const f32x4 v  = V4[i];
    const f32x4 t  = T4[i];
    const f32x4 a  = A4[n4];                            // 128 KB, L2-resident
    const f32x4 am = AM4[n4];

    f32x4 x, nt;
#pragma unroll
    for (int c = 0; c < 4; ++c) {
        // Match reference rounding exactly: mthr = V - (thr + 1); X = mthr >= 0
        const float mthr = __fsub_rn(v[c], __fadd_rn(t[c], 1.0f));
        const float xs   = (mthr >= 0.0f) ? 1.0f : 0.0f;
        x[c]  = xs;
        // new_thr = thr*alpha + X*amp  (mul, mul, add -- no FMA contraction)
        nt[c] = __fadd_rn(__fmul_rn(t[c], a[c]), __fmul_rn(xs, am[c]));
    }

    // Slice 0: X  (outputs are streamed once -> non-temporal stores)
    __builtin_nontemporal_store(x, OUT4 + i);
    // Last slice: new_threshold
    __builtin_nontemporal_store(nt, OUT4 + (size_t)(1 + nd + ndx) * BN4 + i);

    // Delayed-slice copies (pure streaming memcpy per slice).
#pragma unroll
    for (int k = 0; k < nd + ndx; ++k) {
        const int d = __builtin_amdgcn_readfirstlane(
            (k < nd) ? delays[k] : delays_x[k - nd]);
        f32x4 s;
        if (d == 0) {
            s = x;                                      // load skipped wave-wide
        } else {
            s = BUF4[(size_t)(d - 1) * BN4 + i];
        }
        __builtin_nontemporal_store(s, OUT4 + (size_t)(1 + k) * BN4 + i);
    }
}

// Scalar fallback: handles N % 4 != 0 (whole problem) or the flat tail.
__global__ __launch_bounds__(BLOCK)
void alif_scalar_kernel(const float* __restrict__ V,
                        const float* __restrict__ T,
                        const float* __restrict__ alpha,
                        const float* __restrict__ amp,
                        const float* __restrict__ buf,
                        const int*   __restrict__ delays,
                        const int*   __restrict__ delays_x,
                        float*       __restrict__ out,
                        int N, int BN, int start, int nd, int ndx)
{
    const int i = start + blockIdx.x * BLOCK + threadIdx.x;
    if (i >= BN) return;
    const int n = i % N;
    const float v = V[i], t = T[i];
    const float mthr = __fsub_rn(v, __fadd_rn(t, 1.0f));
    const float x = (mthr >= 0.0f) ? 1.0f : 0.0f;
    const float nt = __fadd_rn(__fmul_rn(t, alpha[n]), __fmul_rn(x, amp[n]));
    out[i] = x;
    out[(size_t)(1 + nd + ndx) * BN + i] = nt;
    for (int k = 0; k < nd + ndx; ++k) {
        const int d = __builtin_amdgcn_readfirstlane(
            (k < nd) ? delays[k] : delays_x[k - nd]);
        out[(size_t)(1 + k) * BN + i] =
            (d == 0) ? x : buf[(size_t)(d - 1) * BN + i];
    }
}

extern "C" void kernel_launch(void* const* d_in, const int* in_sizes, int n_in,
                              void* d_out, int out_size, void* d_ws, size_t ws_size,
                              hipStream_t stream)
{
    const float* V        = (const float*)d_in[0];   // (B, N)
    const float* thr      = (const float*)d_in[1];   // (B, N)
    const float* alpha    = (const float*)d_in[2];   // (N,)
    const float* amp      = (const float*)d_in[3];   // (N,)
    const float* buffer   = (const float*)d_in[4];   // (DMAX, B, N)
    const int*   delays   = (const int*)d_in[5];     // (ND,)
    const int*   delays_x = (const int*)d_in[6];     // (NDX,)
    float*       out      = (float*)d_out;           // (1+ND+NDX+1, B, N)

    const int BN  = in_sizes[0];
    const int N   = in_sizes[2];
    const int ND  = in_sizes[5];
    const int NDX = in_sizes[6];
    const int B   = (N > 0) ? (BN / N) : 0;

    if (N > 0 && (N % 4) == 0 && B > 0) {
        const int N4  = N / 4;
        const int BN4 = BN / 4;
        dim3 grid((N4 + BLOCK - 1) / BLOCK, B);
        if (ND == 8 && NDX == 4) {
            alif_fused_kernel<8, 4><<<grid, BLOCK, 0, stream>>>(
                (const f32x4*)V, (const f32x4*)thr, (const f32x4*)alpha,
                (const f32x4*)amp, (const f32x4*)buffer, delays, delays_x,
                (f32x4*)out, N4, BN4, ND, NDX);
        } else {
            alif_fused_kernel<-1, -1><<<grid, BLOCK, 0, stream>>>(
                (const f32x4*)V, (const f32x4*)thr, (const f32x4*)alpha,
                (const f32x4*)amp, (const f32x4*)buffer, delays, delays_x,
                (f32x4*)out, N4, BN4, ND, NDX);
        }
    } else if (BN > 0) {
        const int blocks = (BN + BLOCK - 1) / BLOCK;
        alif_scalar_kernel<<<blocks, BLOCK, 0, stream>>>(
            V, thr, alpha, amp, buffer, delays, delays_x, out,
            N, BN, 0, ND, NDX);
    }
}